// SparseCrossAttention_70068096467032
// MI455X (gfx1250) — compile-verified
//
#include <hip/hip_runtime.h>

#define Bb  4
#define Ss  1024
#define Pp  1024
#define Dd  512
#define Hh  8
#define DHd 64
#define CHUNK 128

typedef float v2f __attribute__((ext_vector_type(2)));
typedef float v8f __attribute__((ext_vector_type(8)));

// ---------------------------------------------------------------------------
// GEMM: Y[m][n] = sum_d X[m][d] * W[n][d] + bias[n]   (M=4096, N=512, K=512)
// One wave per 16x64 tile (4 N-subtiles reuse each A fragment).
// A (16x4 f32): lanes 0-15 rows, vgpr pair = K{0,1}; lanes 16-31 = K{2,3}.
// B (4x16 f32): lane n holds column n; same K split. B[k][n] = W[n][k].
// D (16x16 f32): lane = n + 16*(m>=8); vgpr v = m%8.
// ---------------------------------------------------------------------------
__global__ __launch_bounds__(32)
void gemm_xwT_bias(const float* __restrict__ X, const float* __restrict__ W,
                   const float* __restrict__ bias, float* __restrict__ Y)
{
    const int lane   = threadIdx.x;
    const int laneLo = lane & 15;
    const int sel    = lane >> 4;            // 0 -> K{0,1}, 1 -> K{2,3}
    const int rowBase = blockIdx.x * 16;
    const int colBase = blockIdx.y * 64;

    const float* arow = X + (size_t)(rowBase + laneLo) * Dd + 2 * sel;
    const float* brow = W + (size_t)(colBase + laneLo) * Dd + 2 * sel;

    v8f acc[4] = {v8f{}, v8f{}, v8f{}, v8f{}};

#pragma unroll 4
    for (int k0 = 0; k0 < Dd; k0 += 4) {
        v2f a = *(const v2f*)(arow + k0);
#pragma unroll
        for (int nt = 0; nt < 4; ++nt) {
            v2f b = *(const v2f*)(brow + (size_t)nt * 16 * Dd + k0);
            acc[nt] = __builtin_amdgcn_wmma_f32_16x16x4_f32(
                false, a, false, b, (short)0, acc[nt], false, false);
        }
    }

    const int m0 = rowBase + sel * 8;
#pragma unroll
    for (int nt = 0; nt < 4; ++nt) {
        const int n = colBase + nt * 16 + laneLo;
        const float bn = bias[n];
#pragma unroll
        for (int v = 0; v < 8; ++v) {
            Y[(size_t)(m0 + v) * Dd + n] = acc[nt][v] + bn;
        }
    }
}

// ---------------------------------------------------------------------------
// Per-segment sparse attention with chunked online softmax.
// One 256-thread block per segment (b,s). Edges sorted by (b,s,p): binary
// search gives the contiguous range. Output written as O[b][h][s][dh].
// ---------------------------------------------------------------------------
__global__ __launch_bounds__(256)
void edge_attn(const float* __restrict__ Q,  const float* __restrict__ Km,
               const float* __restrict__ Vm, const int* __restrict__ bI,
               const int* __restrict__ sI,   const int* __restrict__ pI,
               float* __restrict__ O, int E)
{
    __shared__ float qs[Dd + (Dd >> 4)];      // q row, padded: idx = d + (d>>4)
    __shared__ float lw[CHUNK][Hh];           // logits -> weights
    __shared__ float mh[Hh], sh[Hh], rh[Hh];  // running max / sum / rescale
    __shared__ int   erange[2];

    const int t   = threadIdx.x;
    const int seg = blockIdx.x;
    const int b   = seg >> 10;                // seg = b*S + s, S=1024

    if (t == 0) {
        int lo = 0, hi = E;
        while (lo < hi) { int mid = (lo + hi) >> 1;
            int key = bI[mid] * Ss + sI[mid];
            if (key < seg) lo = mid + 1; else hi = mid; }
        erange[0] = lo;
        hi = E;
        while (lo < hi) { int mid = (lo + hi) >> 1;
            int key = bI[mid] * Ss + sI[mid];
            if (key < seg + 1) lo = mid + 1; else hi = mid; }
        erange[1] = lo;
    }
    if (t < Hh) { mh[t] = -3.0e38f; sh[t] = 0.0f; }
    {
        const float* qrow = Q + (size_t)seg * Dd;
        int d0 = t, d1 = t + 256;
        qs[d0 + (d0 >> 4)] = qrow[d0];
        qs[d1 + (d1 >> 4)] = qrow[d1];
    }
    __syncthreads();

    const int e0 = erange[0], e1 = erange[1];
    const int wave = t >> 5, lane = t & 31;
    const int h0 = t >> 6, dh = t & 63;       // outputs (h0, dh) and (h0+4, dh)
    const int h1 = h0 + 4;
    float acc0 = 0.0f, acc1 = 0.0f;

    for (int ec = e0; ec < e1; ec += CHUNK) {
        const int n = min(CHUNK, e1 - ec);

        // Phase 1: logits for this chunk (one edge per wave iteration)
        for (int i = wave; i < n; i += 8) {
            const int p = pI[ec + i];
            const float* krow = Km + (size_t)(b * Pp + p) * Dd + lane * 16;
            const float* qsl  = qs + lane * 16 + lane;   // padded base
            float part = 0.0f;
#pragma unroll
            for (int j = 0; j < 16; j += 4) {
                float4 kv = *(const float4*)(krow + j);
                part += qsl[j]     * kv.x + qsl[j + 1] * kv.y
                      + qsl[j + 2] * kv.z + qsl[j + 3] * kv.w;
            }
            // reduce the 4 lanes that share one head (64 d-values)
            part += __shfl_xor(part, 1, 32);
            part += __shfl_xor(part, 2, 32);
            if ((lane & 3) == 0) lw[i][lane >> 2] = part * 0.125f;  // 1/sqrt(64)
        }
        __syncthreads();

        // Phase 2: per-head online max/sum update
        if (t < Hh) {
            float mOld = mh[t], mx = mOld;
            for (int i = 0; i < n; ++i) mx = fmaxf(mx, lw[i][t]);
            float r  = __expf(mOld - mx);
            float ss = sh[t] * r;
            for (int i = 0; i < n; ++i) {
                float w = __expf(lw[i][t] - mx);
                lw[i][t] = w;
                ss += w;
            }
            mh[t] = mx; sh[t] = ss; rh[t] = r;
        }
        __syncthreads();

        // Phase 3: rescale accumulators, add weighted V
        {
            acc0 *= rh[h0];
            acc1 *= rh[h1];
            for (int i = 0; i < n; ++i) {
                const int p = pI[ec + i];
                const float* vrow = Vm + (size_t)(b * Pp + p) * Dd;
                acc0 += lw[i][h0] * vrow[h0 * DHd + dh];
                acc1 += lw[i][h1] * vrow[h1 * DHd + dh];
            }
        }
        __syncthreads();
    }

    const float o0 = acc0 / (sh[h0] + 1e-9f);
    const float o1 = acc1 / (sh[h1] + 1e-9f);
    const int s = seg & (Ss - 1);
    O[(((size_t)b * Hh + h0) * Ss + s) * DHd + dh] = o0;
    O[(((size_t)b * Hh + h1) * Ss + s) * DHd + dh] = o1;
}

// ---------------------------------------------------------------------------
extern "C" void kernel_launch(void* const* d_in, const int* in_sizes, int n_in,
                              void* d_out, int out_size, void* d_ws, size_t ws_size,
                              hipStream_t stream)
{
    const float* shelf   = (const float*)d_in[0];
    const float* product = (const float*)d_in[1];
    const float* Wq = (const float*)d_in[2];
    const float* bq = (const float*)d_in[3];
    const float* Wk = (const float*)d_in[4];
    const float* bk = (const float*)d_in[5];
    const float* Wv = (const float*)d_in[6];
    const float* bv = (const float*)d_in[7];
    const float* Wo = (const float*)d_in[8];
    const float* bo = (const float*)d_in[9];
    const int*   bI = (const int*)d_in[10];
    const int*   sI = (const int*)d_in[11];
    const int*   pI = (const int*)d_in[12];
    const int    E  = in_sizes[10];

    float* Qbuf = (float*)d_ws;                          // [B*S, 512]
    float* Kbuf = Qbuf + (size_t)Bb * Ss * Dd;           // [B*P, 512]
    float* Vbuf = Kbuf + (size_t)Bb * Pp * Dd;           // [B*P, 512]
    float* Obuf = Vbuf + (size_t)Bb * Pp * Dd;           // [B,H,S,DH] == flat [B*S,512]
    float* Y    = (float*)d_out;

    dim3 gGrid(Bb * Ss / 16, Dd / 64), gBlk(32);
    hipLaunchKernelGGL(gemm_xwT_bias, gGrid, gBlk, 0, stream, shelf,   Wq, bq, Qbuf);
    hipLaunchKernelGGL(gemm_xwT_bias, gGrid, gBlk, 0, stream, product, Wk, bk, Kbuf);
    hipLaunchKernelGGL(gemm_xwT_bias, gGrid, gBlk, 0, stream, product, Wv, bv, Vbuf);
    hipLaunchKernelGGL(edge_attn, dim3(Bb * Ss), dim3(256), 0, stream,
                       Qbuf, Kbuf, Vbuf, bI, sI, pI, Obuf, E);
    hipLaunchKernelGGL(gemm_xwT_bias, gGrid, gBlk, 0, stream, Obuf, Wo, bo, Y);
}